// MoELayer_6313601925508
// MI455X (gfx1250) — compile-verified
//
#include <hip/hip_runtime.h>
#include <math.h>
#include <stdint.h>

// ---------------- problem constants (from reference setup) ----------------
#define N_TOK 16384          // B*T = 4*4096
#define DDIM  1024
#define FDIM  4096
#define NEXP  8
#define CAPTY 2561           // int(N/E * 1.25) + 1
#define CP    2624           // capacity padded to 41*64 tiles

typedef __attribute__((ext_vector_type(2))) float v2f;
typedef __attribute__((ext_vector_type(8))) float v8f;

// ---- gfx1250 async Global->LDS staging (ASYNCcnt path), with safe fallback
#if defined(__has_builtin)
#if __has_builtin(__builtin_amdgcn_global_load_async_to_lds_b128) && \
    __has_builtin(__builtin_amdgcn_s_wait_asynccnt)
#define ASYNC_COPY 1
#endif
#endif
#ifndef ASYNC_COPY
#define ASYNC_COPY 0
#endif

#if ASYNC_COPY
// prototype (from compiler diagnostic): arg0 = AS1 "int __vector(4)" pointer
typedef int v4i __attribute__((__vector_size__(4 * sizeof(int))));
typedef __attribute__((address_space(1))) v4i gv4i_t;
typedef __attribute__((address_space(3))) v4i lv4i_t;
__device__ __forceinline__ void cp16(const void* g, void* l) {
  // flat LDS pointer keeps the LDS offset in addr[31:0] (ISA 10.2)
  __builtin_amdgcn_global_load_async_to_lds_b128(
      (gv4i_t*)(uintptr_t)g, (lv4i_t*)(uint32_t)(uintptr_t)l, 0, 0);
}
__device__ __forceinline__ void cp_drain() {
  __builtin_amdgcn_s_wait_asynccnt(0);
}
#else
__device__ __forceinline__ void cp16(const void* g, void* l) {
  *(uint4*)l = *(const uint4*)g;
}
__device__ __forceinline__ void cp_drain() {}
#endif

__device__ __forceinline__ float gelu_erf(float x) {
  return 0.5f * x * (1.0f + erff(x * 0.70710678118654752f));
}

// ---------------- 1) router: logits -> softmax -> top1 + combine ----------
__global__ __launch_bounds__(256) void router_kernel(
    const float* __restrict__ x, const float* __restrict__ wg,
    int* __restrict__ top1, float* __restrict__ combine) {
  const int wid  = threadIdx.x >> 5;
  const int lane = threadIdx.x & 31;
  const int t = blockIdx.x * 8 + wid;           // one wave32 per token
  float acc[NEXP];
#pragma unroll
  for (int e = 0; e < NEXP; ++e) acc[e] = 0.0f;
  const float* xr = x + (size_t)t * DDIM;
  for (int d = lane; d < DDIM; d += 32) {
    float xv = xr[d];
    const float* wr = wg + (size_t)d * NEXP;
#pragma unroll
    for (int e = 0; e < NEXP; ++e) acc[e] += xv * wr[e];
  }
#pragma unroll
  for (int e = 0; e < NEXP; ++e) {
#pragma unroll
    for (int off = 16; off > 0; off >>= 1)
      acc[e] += __shfl_xor(acc[e], off, 32);
  }
  float m = acc[0]; int am = 0;
#pragma unroll
  for (int e = 1; e < NEXP; ++e) if (acc[e] > m) { m = acc[e]; am = e; }
  float s = 0.0f;
#pragma unroll
  for (int e = 0; e < NEXP; ++e) s += expf(acc[e] - m);
  if (lane == 0) { top1[t] = am; combine[t] = 1.0f / s; }  // softmax at argmax
}

// ---------------- 2) deterministic rank/slot assignment -------------------
__global__ __launch_bounds__(256) void count_kernel(
    const int* __restrict__ top1, int* __restrict__ blkCnt) {
  __shared__ int cnt[NEXP];
  if (threadIdx.x < NEXP) cnt[threadIdx.x] = 0;
  __syncthreads();
  atomicAdd(&cnt[top1[blockIdx.x * 256 + threadIdx.x]], 1);
  __syncthreads();
  if (threadIdx.x < NEXP) blkCnt[blockIdx.x * NEXP + threadIdx.x] = cnt[threadIdx.x];
}

__global__ void scan_kernel(const int* __restrict__ blkCnt, int* __restrict__ blkOff) {
  int e = threadIdx.x;
  if (e >= NEXP) return;
  int run = 0;
  for (int b = 0; b < N_TOK / 256; ++b) {
    blkOff[b * NEXP + e] = run;
    run += blkCnt[b * NEXP + e];
  }
}

__global__ __launch_bounds__(256) void slot_kernel(
    const int* __restrict__ top1, const int* __restrict__ blkOff,
    int* __restrict__ slot, int* __restrict__ tokOfSlot) {
  __shared__ int sh[256];
  const int t = blockIdx.x * 256 + threadIdx.x;
  const int e = top1[t];
  sh[threadIdx.x] = e;
  __syncthreads();
  int local = 0;
  for (int j = 0; j < (int)threadIdx.x; ++j) local += (sh[j] == e) ? 1 : 0;
  const int rank = blkOff[blockIdx.x * NEXP + e] + local;
  const int s = (rank < CAPTY) ? (e * CP + rank) : -1;
  slot[t] = s;
  if (s >= 0) tokOfSlot[s] = t;
}

__global__ void init_toks(int* __restrict__ p, int n) {
  int i = blockIdx.x * 256 + threadIdx.x;
  if (i < n) p[i] = -1;
}

// zero only rows of dropped tokens (GEMM2 writes every valid row)
__global__ __launch_bounds__(256) void zero_dropped(
    const int* __restrict__ slot, float* __restrict__ out) {
  if (slot[blockIdx.x] >= 0) return;
  float4* dst = (float4*)(out + (size_t)blockIdx.x * DDIM);
  dst[threadIdx.x] = make_float4(0.f, 0.f, 0.f, 0.f);
}

// ---------------- 3) GEMM1: H = gelu(gather(x) @ W1 + b1), H in f16 -------
// 64x64 block tile, 4 waves, wave does 32x32 via 2x2 V_WMMA_F32_16X16X4_F32.
__global__ __launch_bounds__(128) void moe_gemm1(
    const float* __restrict__ x,        // [N_TOK, DDIM]
    const float* __restrict__ W1,       // [E, DDIM, FDIM]
    const float* __restrict__ b1,       // [E, FDIM]
    const int* __restrict__ tokOfSlot,  // [E*CP]
    _Float16* __restrict__ H) {         // [E, CP, FDIM]
  __shared__ float As[64][20];          // padded rows (80B, 16B aligned)
  __shared__ float Bs[16][68];          // padded rows (272B, 16B aligned)
  __shared__ int shTok[64];

  const int tid = threadIdx.x;
  const int wid = tid >> 5, lane = tid & 31, lo = lane & 15, hi = lane >> 4;
  const int wm = (wid >> 1) * 32, wn = (wid & 1) * 32;
  const int e = blockIdx.z, row0 = blockIdx.y * 64, col0 = blockIdx.x * 64;

  if (tid < 64) {
    int t0 = tokOfSlot[e * CP + row0 + tid];
    shTok[tid] = (t0 < 0) ? 0 : t0;     // clamp empty slots (rows never used)
  }
  __syncthreads();

  const float* Wb = W1 + (size_t)e * DDIM * FDIM + col0;
  const int ra = tid >> 2, ca = (tid & 3) * 4;    // A: rows ra, ra+32
  const int rb = tid >> 4, cb = (tid & 15) * 4;   // B: rows rb, rb+8
  const size_t arow0 = (size_t)shTok[ra] * DDIM;
  const size_t arow1 = (size_t)shTok[ra + 32] * DDIM;

  v8f acc[2][2] = {};
  for (int kk = 0; kk < DDIM; kk += 16) {
    cp16(x + arow0 + kk + ca, &As[ra][ca]);               // gathered A rows
    cp16(x + arow1 + kk + ca, &As[ra + 32][ca]);
    cp16(Wb + (size_t)(kk + rb) * FDIM + cb, &Bs[rb][cb]);
    cp16(Wb + (size_t)(kk + rb + 8) * FDIM + cb, &Bs[rb + 8][cb]);
    cp_drain();
    __syncthreads();
#pragma unroll
    for (int ks = 0; ks < 4; ++ks) {
      const int kb = ks * 4 + 2 * hi;   // lanes 0-15: K=0,1 ; 16-31: K=2,3
      v2f a0 = *(const v2f*)&As[wm + lo][kb];
      v2f a1 = *(const v2f*)&As[wm + 16 + lo][kb];
      v2f b0, b1v;
      b0.x  = Bs[kb][wn + lo];       b0.y  = Bs[kb + 1][wn + lo];
      b1v.x = Bs[kb][wn + 16 + lo];  b1v.y = Bs[kb + 1][wn + 16 + lo];
      acc[0][0] = __builtin_amdgcn_wmma_f32_16x16x4_f32(false, a0, false, b0,
                      (short)0, acc[0][0], false, false);
      acc[0][1] = __builtin_amdgcn_wmma_f32_16x16x4_f32(false, a0, false, b1v,
                      (short)0, acc[0][1], false, false);
      acc[1][0] = __builtin_amdgcn_wmma_f32_16x16x4_f32(false, a1, false, b0,
                      (short)0, acc[1][0], false, false);
      acc[1][1] = __builtin_amdgcn_wmma_f32_16x16x4_f32(false, a1, false, b1v,
                      (short)0, acc[1][1], false, false);
    }
    __syncthreads();
  }

  _Float16* Hb = H + (size_t)e * CP * FDIM;
#pragma unroll
  for (int i = 0; i < 2; ++i) {
#pragma unroll
    for (int j = 0; j < 2; ++j) {
      const int ncol = col0 + wn + j * 16 + lo;
      const float bv = b1[e * FDIM + ncol];
      const int mbase = row0 + wm + i * 16 + hi * 8;
#pragma unroll
      for (int r = 0; r < 8; ++r) {
        Hb[(size_t)(mbase + r) * FDIM + ncol] =
            (_Float16)gelu_erf(acc[i][j][r] + bv);
      }
    }
  }
}

// ---------------- 4) GEMM2: out[tok] = (H @ W2 + b2) * combine ------------
__global__ __launch_bounds__(128) void moe_gemm2(
    const _Float16* __restrict__ H,     // [E, CP, FDIM]
    const float* __restrict__ W2,       // [E, FDIM, DDIM]
    const float* __restrict__ b2,       // [E, DDIM]
    const int* __restrict__ tokOfSlot,
    const float* __restrict__ combine,
    float* __restrict__ out) {          // [N_TOK, DDIM]
  __shared__ _Float16 Ah[64][24];       // padded rows (48B, 16B aligned)
  __shared__ float Bs[16][68];

  const int tid = threadIdx.x;
  const int wid = tid >> 5, lane = tid & 31, lo = lane & 15, hi = lane >> 4;
  const int wm = (wid >> 1) * 32, wn = (wid & 1) * 32;
  const int e = blockIdx.z, row0 = blockIdx.y * 64, col0 = blockIdx.x * 64;

  const _Float16* Ab = H + (size_t)e * CP * FDIM + (size_t)row0 * FDIM;
  const float* Wb = W2 + (size_t)e * FDIM * DDIM + col0;
  const int ra = tid >> 1, ca = (tid & 1) * 8;    // A: 64 rows x 16 halves
  const int rb = tid >> 4, cb = (tid & 15) * 4;

  v8f acc[2][2] = {};
  for (int kk = 0; kk < FDIM; kk += 16) {
    cp16(Ab + (size_t)ra * FDIM + kk + ca, &Ah[ra][ca]);
    cp16(Wb + (size_t)(kk + rb) * DDIM + cb, &Bs[rb][cb]);
    cp16(Wb + (size_t)(kk + rb + 8) * DDIM + cb, &Bs[rb + 8][cb]);
    cp_drain();
    __syncthreads();
#pragma unroll
    for (int ks = 0; ks < 4; ++ks) {
      const int kb = ks * 4 + 2 * hi;
      v2f a0, a1;
      {
        union { unsigned u; _Float16 h[2]; } t0, t1;
        t0.u = *(const unsigned*)&Ah[wm + lo][kb];
        t1.u = *(const unsigned*)&Ah[wm + 16 + lo][kb];
        a0.x = (float)t0.h[0]; a0.y = (float)t0.h[1];
        a1.x = (float)t1.h[0]; a1.y = (float)t1.h[1];
      }
      v2f b0, b1v;
      b0.x  = Bs[kb][wn + lo];       b0.y  = Bs[kb + 1][wn + lo];
      b1v.x = Bs[kb][wn + 16 + lo];  b1v.y = Bs[kb + 1][wn + 16 + lo];
      acc[0][0] = __builtin_amdgcn_wmma_f32_16x16x4_f32(false, a0, false, b0,
                      (short)0, acc[0][0], false, false);
      acc[0][1] = __builtin_amdgcn_wmma_f32_16x16x4_f32(false, a0, false, b1v,
                      (short)0, acc[0][1], false, false);
      acc[1][0] = __builtin_amdgcn_wmma_f32_16x16x4_f32(false, a1, false, b0,
                      (short)0, acc[1][0], false, false);
      acc[1][1] = __builtin_amdgcn_wmma_f32_16x16x4_f32(false, a1, false, b1v,
                      (short)0, acc[1][1], false, false);
    }
    __syncthreads();
  }

#pragma unroll
  for (int i = 0; i < 2; ++i) {
#pragma unroll
    for (int j = 0; j < 2; ++j) {
      const int ncol = col0 + wn + j * 16 + lo;
      const float bv = b2[e * DDIM + ncol];
      const int mbase = row0 + wm + i * 16 + hi * 8;
#pragma unroll
      for (int r = 0; r < 8; ++r) {
        const int tok = tokOfSlot[e * CP + mbase + r];
        if (tok >= 0)
          out[(size_t)tok * DDIM + ncol] = (acc[i][j][r] + bv) * combine[tok];
      }
    }
  }
}

// ---------------- launch ---------------------------------------------------
extern "C" void kernel_launch(void* const* d_in, const int* in_sizes, int n_in,
                              void* d_out, int out_size, void* d_ws, size_t ws_size,
                              hipStream_t stream) {
  const float* x  = (const float*)d_in[0];
  const float* wg = (const float*)d_in[1];
  const float* W1 = (const float*)d_in[2];
  const float* b1 = (const float*)d_in[3];
  const float* W2 = (const float*)d_in[4];
  const float* b2 = (const float*)d_in[5];
  float* out = (float*)d_out;

  char* ws = (char*)d_ws;
  size_t off = 0;
  auto alloc = [&](size_t bytes) -> void* {
    void* p = ws + off;
    off += bytes;
    off = (off + 255) & ~(size_t)255;
    return p;
  };
  int*       top1   = (int*)      alloc((size_t)N_TOK * 4);
  int*       slot   = (int*)      alloc((size_t)N_TOK * 4);
  float*     comb   = (float*)    alloc((size_t)N_TOK * 4);
  int*       blkCnt = (int*)      alloc((size_t)(N_TOK / 256) * NEXP * 4);
  int*       blkOff = (int*)      alloc((size_t)(N_TOK / 256) * NEXP * 4);
  int*       tokOf  = (int*)      alloc((size_t)NEXP * CP * 4);
  _Float16*  H      = (_Float16*) alloc((size_t)NEXP * CP * FDIM * 2);  // 172 MB

  router_kernel<<<N_TOK / 8,   256, 0, stream>>>(x, wg, top1, comb);
  count_kernel <<<N_TOK / 256, 256, 0, stream>>>(top1, blkCnt);
  scan_kernel  <<<1, 32, 0, stream>>>(blkCnt, blkOff);
  init_toks    <<<(NEXP * CP + 255) / 256, 256, 0, stream>>>(tokOf, NEXP * CP);
  slot_kernel  <<<N_TOK / 256, 256, 0, stream>>>(top1, blkOff, slot, tokOf);
  zero_dropped <<<N_TOK, 256, 0, stream>>>(slot, out);

  moe_gemm1<<<dim3(FDIM / 64, CP / 64, NEXP), 128, 0, stream>>>(
      x, W1, b1, tokOf, H);
  moe_gemm2<<<dim3(DDIM / 64, CP / 64, NEXP), 128, 0, stream>>>(
      H, W2, b2, tokOf, comb, out);
}